// GaussianSplatting2D_54348516164089
// MI455X (gfx1250) — compile-verified
//
#include <hip/hip_runtime.h>
#include <math.h>

#define NG 1024
#define WI 256
#define HI 256
#define EPS2D 0.3f

typedef __attribute__((ext_vector_type(2))) float v2f;
typedef __attribute__((ext_vector_type(8))) float v8f;

// ---------------------------------------------------------------------------
// Phase 1: per-gaussian preprocessing + stable O(N^2) rank sort by depth z.
// One block of 1024 threads (32 waves). Writes 7 sorted arrays (28KB) to ws:
//   [u | v | conic_a | conic_b | conic_c | opacity | color], each NG floats.
// ---------------------------------------------------------------------------
__global__ __launch_bounds__(1024) void gs_prep(
    const float* __restrict__ means, const float* __restrict__ quats,
    const float* __restrict__ scales, const float* __restrict__ rgbs,
    const float* __restrict__ opac, const float* __restrict__ vm,
    const float* __restrict__ Km, float* __restrict__ ws) {
  __shared__ float zsh[NG];
  const int i = threadIdx.x;

  const float R[3][3] = {{vm[0], vm[1], vm[2]},
                         {vm[4], vm[5], vm[6]},
                         {vm[8], vm[9], vm[10]}};
  const float t0 = vm[3], t1 = vm[7], t2 = vm[11];
  const float fx = Km[0], cx = Km[2], fy = Km[4], cy = Km[5];

  const float mx = means[3 * i], my = means[3 * i + 1], mz = means[3 * i + 2];
  const float x = R[0][0] * mx + R[0][1] * my + R[0][2] * mz + t0;
  const float y = R[1][0] * mx + R[1][1] * my + R[1][2] * mz + t1;
  const float z = R[2][0] * mx + R[2][1] * my + R[2][2] * mz + t2;
  const float invz = 1.0f / z;
  const float u = fx * x * invz + cx;
  const float v = fy * y * invz + cy;

  const float q = quats[i];
  const float ct = cosf(q), st = sinf(q);
  const float sx = scales[3 * i], sy = scales[3 * i + 1], sz = scales[3 * i + 2];
  // cov3 = (R3 * diag-ish scales) (R3 * scales)^T  -- symmetric
  const float C00 = ct * ct * sx * sx + st * st * sy * sy;
  const float C01 = ct * st * sx * sx - st * ct * sy * sy;
  const float C11 = st * st * sx * sx + ct * ct * sy * sy;
  const float C22 = sz * sz;
  float cov3[3][3] = {{C00, C01, 0.0f}, {C01, C11, 0.0f}, {0.0f, 0.0f, C22}};

  // cov_cam = R * cov3 * R^T
  float tmp[3][3], Vc[3][3];
#pragma unroll
  for (int r = 0; r < 3; ++r)
#pragma unroll
    for (int c = 0; c < 3; ++c)
      tmp[r][c] = R[r][0] * cov3[0][c] + R[r][1] * cov3[1][c] + R[r][2] * cov3[2][c];
#pragma unroll
  for (int r = 0; r < 3; ++r)
#pragma unroll
    for (int c = 0; c < 3; ++c)
      Vc[r][c] = tmp[r][0] * R[c][0] + tmp[r][1] * R[c][1] + tmp[r][2] * R[c][2];

  // Jacobian rows
  const float j0[3] = {fx * invz, 0.0f, -fx * x * invz * invz};
  const float j1[3] = {0.0f, fy * invz, -fy * y * invz * invz};
  float a = 0.0f, b = 0.0f, c2 = 0.0f;
#pragma unroll
  for (int p = 0; p < 3; ++p)
#pragma unroll
    for (int r = 0; r < 3; ++r) {
      a += j0[p] * Vc[p][r] * j0[r];
      b += j0[p] * Vc[p][r] * j1[r];
      c2 += j1[p] * Vc[p][r] * j1[r];
    }
  a += EPS2D;
  c2 += EPS2D;
  const float det = a * c2 - b * b;
  const float idet = 1.0f / det;
  const float ica = c2 * idet, icb = -b * idet, icc = a * idet;

  const float op = 1.0f / (1.0f + expf(-opac[i]));
  const float colr = 1.0f / (1.0f + expf(-rgbs[i]));

  zsh[i] = z;
  __syncthreads();

  // stable rank (matches jnp.argsort): ties broken by original index
  int rank = 0;
  for (int j = 0; j < NG; ++j) {
    const float zj = zsh[j];
    rank += (zj < z) || (zj == z && j < i);
  }
  ws[rank] = u;
  ws[NG + rank] = v;
  ws[2 * NG + rank] = ica;
  ws[3 * NG + rank] = icb;
  ws[4 * NG + rank] = icc;
  ws[5 * NG + rank] = op;
  ws[6 * NG + rank] = colr;
}

// ---------------------------------------------------------------------------
// Phase 2: render. One wave == one 16x1 pixel strip. For each chunk of 16
// depth-sorted gaussians, a single V_WMMA_F32_16X16X4_F32 evaluates sigma for
// 16 gaussians x 16 pixels:
//   A (16x4): per-gaussian tile coeffs [0.5*ca, ca*du+cb*dv, const, 0]
//   B (4x16): per-pixel basis          [qx^2,   qx,          1,     0]
// D layout: lane holds pixel (lane%16), gaussian rows 8*(lane/16)+i -> the
// alpha-compositing scan is lane-local over 8 ordered gaussians; the two
// half-wave segments merge associatively via one shfl_xor(16).
// ---------------------------------------------------------------------------
__global__ __launch_bounds__(256) void gs_render(const float* __restrict__ ws,
                                                 float* __restrict__ out) {
  __shared__ float sh[7 * NG];  // 28KB sorted gaussian table
  const int tid = threadIdx.x;
  for (int k = tid; k < 7 * NG; k += 256) sh[k] = ws[k];
  __syncthreads();

  const float* su = sh;
  const float* sv = sh + NG;
  const float* sca = sh + 2 * NG;
  const float* scb = sh + 3 * NG;
  const float* scc = sh + 4 * NG;
  const float* sop = sh + 5 * NG;
  const float* scol = sh + 6 * NG;

  const int lane = tid & 31;
  const int wav = tid >> 5;
  const int tile = blockIdx.x * 8 + wav;  // 4096 strips total
  const int tx = tile & 15;               // strip x (16 strips per row)
  const int py = tile >> 4;               // pixel row
  const int col = lane & 15;
  const int half = lane >> 4;

  const float qx = (float)col - 7.5f;      // tile-centered x
  const float xc = (float)(tx * 16) + 8.0f;  // strip center (px coords use +0.5)
  const float yc = (float)py + 0.5f;

  // B matrix: constant for the whole strip
  v2f bm;
  bm.x = half ? 1.0f : qx * qx;  // K=2 : K=0
  bm.y = half ? 0.0f : qx;       // K=3 : K=1

  float Crun = 0.0f, Trun = 1.0f;

  for (int base = 0; base < NG; base += 16) {
    // Per-gaussian, per-strip quadratic coefficients (all 32 lanes: g=base+col)
    const int g = base + col;
    const float u = su[g], v = sv[g];
    const float ca = sca[g], cb = scb[g], cc = scc[g];
    const float du = xc - u;   // dx = qx + du
    const float dv = yc - v;   // dy = dv (constant along strip)
    const float k0 = 0.5f * ca;
    const float k1 = ca * du + cb * dv;
    const float k2 = 0.5f * ca * du * du + 0.5f * cc * dv * dv + cb * du * dv;

    v2f am;
    am.x = half ? k2 : k0;
    am.y = half ? 0.0f : k1;

    v8f sig = {};
    sig = __builtin_amdgcn_wmma_f32_16x16x4_f32(
        /*neg_a=*/false, am, /*neg_b=*/false, bm,
        /*c_mod=*/(short)0, sig, /*reuse_a=*/false, /*reuse_b=*/false);

    // Lane-local alpha compositing over 8 depth-ordered gaussians
    const int gb = base + 8 * half;
    float Cl = 0.0f, Tl = 1.0f;
#pragma unroll
    for (int i2 = 0; i2 < 8; ++i2) {
      float s = fmaxf(sig[i2], 0.0f);
      float al = sop[gb + i2] * __expf(-s);
      al = fminf(al, 0.999f);
      Cl += al * Tl * scol[gb + i2];
      Tl *= 1.0f - al;
    }

    // Merge the two 8-gaussian segments (halves of the wave) in depth order
    const float oC = __shfl_xor(Cl, 16, 32);
    const float oT = __shfl_xor(Tl, 16, 32);
    const float cA = half ? oC : Cl;
    const float tA = half ? oT : Tl;
    const float cB = half ? Cl : oC;
    const float tB = half ? Tl : oT;
    Crun += Trun * (cA + tA * cB);
    Trun *= tA * tB;

    // Wave-uniform early exit once transmittance is dead (error < 1e-6)
    if (__ballot(Trun > 1e-6f) == 0ull) break;
  }

  if (lane < 16) out[py * WI + tx * 16 + col] = Crun;
}

// ---------------------------------------------------------------------------
extern "C" void kernel_launch(void* const* d_in, const int* in_sizes, int n_in,
                              void* d_out, int out_size, void* d_ws, size_t ws_size,
                              hipStream_t stream) {
  const float* means = (const float*)d_in[0];
  const float* quats = (const float*)d_in[1];
  const float* scales = (const float*)d_in[2];
  const float* rgbs = (const float*)d_in[3];
  const float* opac = (const float*)d_in[4];
  const float* viewmat = (const float*)d_in[5];
  const float* Km = (const float*)d_in[6];
  float* out = (float*)d_out;
  float* ws = (float*)d_ws;

  gs_prep<<<1, NG, 0, stream>>>(means, quats, scales, rgbs, opac, viewmat, Km, ws);

  // 4096 strips (16 strips/row * 256 rows), 8 waves per 256-thread block
  gs_render<<<512, 256, 0, stream>>>(ws, out);
}